// QuantumHead_72550587564090
// MI455X (gfx1250) — compile-verified
//
#include <hip/hip_runtime.h>
#include <hip/hip_bf16.h>
#include <math.h>

typedef __attribute__((ext_vector_type(2))) float v2f;
typedef __attribute__((ext_vector_type(8))) float v8f;

#define SEQ_LEN  8192
#define D_EMBED  4096
#define NVOCAB   7
#define NQ       3

__global__ __launch_bounds__(256) void quantum_head_kernel(
    const int*   __restrict__ tok,
    const float* __restrict__ embed,   // (7, 4096) row-major
    const float* __restrict__ W,       // (3, 4096) row-major
    const float* __restrict__ bvec,    // (3,)
    const float* __restrict__ wts,     // (3, 2)
    float*       __restrict__ out)     // (7,)
{
    __shared__ int   counts[8];
    __shared__ float red[8][256];      // 8 waves x (8 VGPR x 32 lane) C tiles
    __shared__ float gsum[256];        // cross-wave reduced C tile

    const int tid = threadIdx.x;
    if (tid < 8) counts[tid] = 0;
    __syncthreads();

    // ---- Phase 1: token histogram (gather-mean collapses to counts) ----
    for (int t = tid; t < SEQ_LEN; t += 256)
        atomicAdd(&counts[tok[t] & 7], 1);

    // ---- Phase 2: G = W(16x4096) x embedT(4096x16) via f32 WMMA ----
    // K split across 8 waves; each wave runs 128 16x16x4 f32 WMMAs (K=512).
    // No zero-padding needed: D[m][n] depends only on A-row m and B-col n,
    // and we only read m<3, n<7. Out-of-range lanes clamp to a valid row,
    // producing harmless garbage in rows/cols of D we never read. This keeps
    // the loop branch-free (no EXEC save/restore around the loads).
    const int wv   = tid >> 5;         // wave id 0..7
    const int lane = tid & 31;
    const int half = lane >> 4;        // 0: K=k0,k0+1  1: K=k0+2,k0+3
    const int r    = lane & 15;        // M row (A) / N col (B)
    const int kb   = wv * (D_EMBED / 8);

    const int rowA = (r < NQ)     ? r : (NQ - 1);
    const int rowB = (r < NVOCAB) ? r : (NVOCAB - 1);
    const float* pa = W     + rowA * D_EMBED + kb + 2 * half;
    const float* pb = embed + rowB * D_EMBED + kb + 2 * half;

    v8f acc = {0.f, 0.f, 0.f, 0.f, 0.f, 0.f, 0.f, 0.f};
#pragma unroll 8
    for (int i = 0; i < (D_EMBED / 8) / 4; ++i) {   // 128 iterations
        v2f a  = *(const v2f*)(pa + 4 * i);
        v2f bm = *(const v2f*)(pb + 4 * i);
        acc = __builtin_amdgcn_wmma_f32_16x16x4_f32(
            /*neg_a=*/false, a, /*neg_b=*/false, bm,
            /*c_mod=*/(short)0, acc, /*reuse_a=*/false, /*reuse_b=*/false);
    }

#pragma unroll
    for (int v = 0; v < 8; ++v)
        red[wv][v * 32 + lane] = acc[v];
    __syncthreads();

    // Parallel cross-wave reduction: each thread reduces one C-tile element
    // across the 8 wave tiles (8 LDS loads + 7 adds, all 256 threads).
    {
        float s = 0.f;
#pragma unroll
        for (int w8 = 0; w8 < 8; ++w8) s += red[w8][tid];
        gsum[tid] = s;
    }
    __syncthreads();

    // ---- Phase 3: tiny tail on one thread (G is 3x7; circuit is 8 amps) ----
    if (tid == 0) {
        // C/D layout: VGPR v, lanes 0..15 -> M=v, N=lane. We need M<3, N<7:
        // G[m][n] = gsum[m*32 + n].
        const float inv_seq = 1.0f / (float)SEQ_LEN;
        float wfrac[NVOCAB];
#pragma unroll
        for (int v = 0; v < NVOCAB; ++v)
            wfrac[v] = (float)counts[v] * inv_seq;

        float ang[NQ];
#pragma unroll
        for (int m = 0; m < NQ; ++m) {
            float s = bvec[m];
#pragma unroll
            for (int v = 0; v < NVOCAB; ++v)
                s += gsum[m * 32 + v] * wfrac[v];
            ang[m] = s;
        }

        // 3-qubit statevector: index = q0*4 + q1*2 + q2
        float sr[8], si[8];
#pragma unroll
        for (int i = 0; i < 8; ++i) { sr[i] = 0.f; si[i] = 0.f; }
        sr[0] = 1.f;
        const int stride[NQ] = {4, 2, 1};

        // RX(theta) then RZ(theta) per qubit (hardware sin/cos: args are tiny)
#pragma unroll
        for (int q = 0; q < NQ; ++q) {
            const float t = 0.5f * ang[q];
            const float c = __cosf(t), s = __sinf(t);
#pragma unroll
            for (int i = 0; i < 8; ++i) {
                if (i & stride[q]) continue;
                const int a = i, b = i + stride[q];
                float ar = sr[a], ai = si[a], br = sr[b], bi = si[b];
                // RX: [[c, -is],[-is, c]]
                sr[a] = c * ar + s * bi;  si[a] = c * ai - s * br;
                sr[b] = c * br + s * ai;  si[b] = c * bi - s * ar;
            }
#pragma unroll
            for (int i = 0; i < 8; ++i) {
                if (i & stride[q]) continue;
                const int a = i, b = i + stride[q];
                float ar = sr[a], ai = si[a], br = sr[b], bi = si[b];
                // RZ: diag(e^{-it}, e^{+it}), t = theta/2
                sr[a] = c * ar + s * ai;  si[a] = c * ai - s * ar;
                sr[b] = c * br - s * bi;  si[b] = c * bi + s * br;
            }
        }

        // CNOT(0,1): q0=1 -> swap q1 : 4<->6, 5<->7
        {
            float tr, ti;
            tr = sr[4]; ti = si[4]; sr[4] = sr[6]; si[4] = si[6]; sr[6] = tr; si[6] = ti;
            tr = sr[5]; ti = si[5]; sr[5] = sr[7]; si[5] = si[7]; sr[7] = tr; si[7] = ti;
        }
        // CNOT(1,2): q1=1 -> swap q2 : 2<->3, 6<->7
        {
            float tr, ti;
            tr = sr[2]; ti = si[2]; sr[2] = sr[3]; si[2] = si[3]; sr[3] = tr; si[3] = ti;
            tr = sr[6]; ti = si[6]; sr[6] = sr[7]; si[6] = si[7]; sr[7] = tr; si[7] = ti;
        }

        // RY(w0*angle + w1) per qubit
#pragma unroll
        for (int q = 0; q < NQ; ++q) {
            const float t = 0.5f * (wts[q * 2 + 0] * ang[q] + wts[q * 2 + 1]);
            const float c = __cosf(t), s = __sinf(t);
#pragma unroll
            for (int i = 0; i < 8; ++i) {
                if (i & stride[q]) continue;
                const int a = i, b = i + stride[q];
                float ar = sr[a], ai = si[a], br = sr[b], bi = si[b];
                sr[a] = c * ar - s * br;  si[a] = c * ai - s * bi;
                sr[b] = s * ar + c * br;  si[b] = s * ai + c * bi;
            }
        }

#pragma unroll
        for (int v = 0; v < NVOCAB; ++v) {
            const float p = sr[v] * sr[v] + si[v] * si[v];
            out[v] = __logf(p + 1e-9f);
        }
    }
}

extern "C" void kernel_launch(void* const* d_in, const int* in_sizes, int n_in,
                              void* d_out, int out_size, void* d_ws, size_t ws_size,
                              hipStream_t stream) {
    const int*   tok   = (const int*)d_in[0];
    const float* embed = (const float*)d_in[1];
    const float* W     = (const float*)d_in[2];
    const float* b     = (const float*)d_in[3];
    const float* wts   = (const float*)d_in[4];
    float*       out   = (float*)d_out;

    hipLaunchKernelGGL(quantum_head_kernel, dim3(1), dim3(256), 0, stream,
                       tok, embed, W, b, wts, out);
}